// MoE_30691836297575
// MI455X (gfx1250) — compile-verified
//
#include <hip/hip_runtime.h>
#include <math.h>

// ---------------------------------------------------------------------------
// Problem constants
// ---------------------------------------------------------------------------
constexpr int TD     = 768;                 // grid dim
constexpr int NB     = 32;                  // batch
constexpr int NE     = 64;                  // experts
constexpr int NF     = 2048;                // frequencies per expert
constexpr int TDTD   = TD * TD;             // 589824
constexpr float ALPHA = 300.0f;
constexpr float OUT_SCALE = ALPHA / (float)TDTD;

// GEMM tiling
constexpr int BT   = 64;                    // block tile (M and N)
constexpr int KT   = 16;                    // k tile
constexpr int LDA  = KT + 4;                // padded LDS stride for A tiles (20 dw -> 80B rows, 16B aligned)
constexpr int LDBP = 2 * BT + 32;           // k-pair-interleaved B stride (160 dw): halves on disjoint banks

typedef float v2f __attribute__((ext_vector_type(2)));
typedef float v8f __attribute__((ext_vector_type(8)));

__device__ __forceinline__ v8f wmma_f32(v2f a, v2f b, v8f c) {
    // D = A(16x4 f32) x B(4x16 f32) + C(16x16 f32)
    return __builtin_amdgcn_wmma_f32_16x16x4_f32(
        /*neg_a=*/false, a, /*neg_b=*/false, b,
        /*c_mod=*/(short)0, c, /*reuse_a=*/false, /*reuse_b=*/false);
}

// Interleave two k-adjacent float4 rows into pair layout and store to LDS.
__device__ __forceinline__ void store_pairs(float* dst, float4 r0, float4 r1) {
    float4 lo, hi;
    lo.x = r0.x; lo.y = r1.x; lo.z = r0.y; lo.w = r1.y;
    hi.x = r0.z; hi.y = r1.z; hi.z = r0.w; hi.w = r1.w;
    *(float4*)(dst)     = lo;
    *(float4*)(dst + 4) = hi;
}

// ---------------------------------------------------------------------------
// 1) Twiddle tables: Tc[a*TD+b] = cos(2*pi*a*b/TD), Ts = sin(...)  (symmetric)
// ---------------------------------------------------------------------------
__global__ __launch_bounds__(256) void twiddle_kernel(float* __restrict__ Tc,
                                                      float* __restrict__ Ts) {
    int t = blockIdx.x * 256 + threadIdx.x;
    if (t >= TDTD) return;
    int a = t / TD;
    int b = t - a * TD;
    int ph = (int)(((long long)a * (long long)b) % TD);   // keep argument small
    float ang = (float)ph * (6.283185307179586476925287f / (float)TD);
    float s, c;
    sincosf(ang, &s, &c);
    Tc[t] = c;
    Ts[t] = s;
}

// ---------------------------------------------------------------------------
// 2) Router: logits = cls @ W^T + b, top-2, renormalized softmax weights.
// ---------------------------------------------------------------------------
__global__ __launch_bounds__(64) void router_kernel(const float* __restrict__ cls,
                                                    const float* __restrict__ W,
                                                    const float* __restrict__ bias,
                                                    int* __restrict__ sel,
                                                    float* __restrict__ selw) {
    const int b = blockIdx.x;
    const int e = threadIdx.x;
    __shared__ float lg[NE];
    const float* x = cls + (size_t)b * 768;
    const float* w = W + (size_t)e * 768;
    float acc = bias[e];
    for (int d = 0; d < 768; ++d) acc += x[d] * w[d];
    lg[e] = acc;
    __syncthreads();
    if (e == 0) {
        int i0 = 0; float l0 = lg[0];
        for (int i = 1; i < NE; ++i) if (lg[i] > l0) { l0 = lg[i]; i0 = i; }
        int i1 = -1; float l1 = -INFINITY;
        for (int i = 0; i < NE; ++i) if (i != i0 && lg[i] > l1) { l1 = lg[i]; i1 = i; }
        float e1 = expf(l1 - l0);                 // top-2 softmax, renormalized
        float w0 = 1.0f / (1.0f + e1);
        sel[b * 2 + 0] = i0;  sel[b * 2 + 1] = i1;
        selw[b * 2 + 0] = w0; selw[b * 2 + 1] = 1.0f - w0;
    }
}

// ---------------------------------------------------------------------------
// 3) Zero the scatter grid (every call: graph replays must not accumulate)
// ---------------------------------------------------------------------------
__global__ __launch_bounds__(256) void zero_kernel(float4* __restrict__ p, long long n4) {
    long long i = (long long)blockIdx.x * 256 + threadIdx.x;
    long long stride = (long long)gridDim.x * 256;
    float4 z; z.x = 0.f; z.y = 0.f; z.z = 0.f; z.w = 0.f;
    for (; i < n4; i += stride) p[i] = z;
}

// ---------------------------------------------------------------------------
// 4) Scatter: indices come from a permutation -> globally unique -> plain store
// ---------------------------------------------------------------------------
__global__ __launch_bounds__(256) void scatter_kernel(const float* __restrict__ coeff,
                                                      const int* __restrict__ lidx,
                                                      const int* __restrict__ sel,
                                                      const float* __restrict__ selw,
                                                      float* __restrict__ mask) {
    int t = blockIdx.x * 256 + threadIdx.x;        // [0, NB*2*NF)
    int b    = t >> 12;                            // 2*NF = 4096 per batch
    int slot = (t >> 11) & 1;
    int f    = t & (NF - 1);
    int e    = sel[b * 2 + slot];
    float w  = selw[b * 2 + slot];
    int p    = lidx[(size_t)e * NF + f];
    float v  = coeff[(size_t)e * NF + f] * w;
    mask[(size_t)b * TDTD + p] = v;
}

// ---------------------------------------------------------------------------
// 5) Stage 1: Xr[b] = mask[b] @ Tc ; Xi[b] = mask[b] @ Ts   (contract cols)
//    64x64 tile, 128 threads (4 waves of 32x32), dual WMMA chains.
//    B tiles in LDS are k-pair interleaved so fragments are single b64 loads.
// ---------------------------------------------------------------------------
__global__ __launch_bounds__(128) void gemm_stage1(const float* __restrict__ mask,
                                                   const float* __restrict__ Tc,
                                                   const float* __restrict__ Ts,
                                                   float* __restrict__ Xr,
                                                   float* __restrict__ Xi) {
    __shared__ float As[BT * LDA];
    __shared__ float Bc[(KT / 2) * LDBP];
    __shared__ float Bs[(KT / 2) * LDBP];

    const int b      = blockIdx.z;
    const int mBlock = blockIdx.y * BT;
    const int nBlock = blockIdx.x * BT;
    const float* Ab  = mask + (size_t)b * TDTD;

    const int tid   = threadIdx.x;
    const int wave  = tid >> 5;
    const int lane  = tid & 31;
    const int waveM = (wave >> 1) * 32;
    const int waveN = (wave & 1) * 32;
    const int lrow  = lane & 15;
    const int lhi   = lane >> 4;

    v8f accR[2][2], accI[2][2];
#pragma unroll
    for (int i = 0; i < 2; ++i)
#pragma unroll
        for (int j = 0; j < 2; ++j) { accR[i][j] = (v8f)(0.0f); accI[i][j] = (v8f)(0.0f); }

    for (int k0 = 0; k0 < TD; k0 += KT) {
        // ---- stage tiles into LDS ----
        {
            int r  = tid >> 2;       // 0..31
            int cg = tid & 3;        // 0..3 (float4 groups across KT=16)
#pragma unroll
            for (int h = 0; h < 2; ++h) {
                int rr = r + 32 * h;
                float4 av = *(const float4*)(Ab + (size_t)(mBlock + rr) * TD + k0 + cg * 4);
                *(float4*)(As + rr * LDA + cg * 4) = av;
            }
            int kp = tid >> 4;       // k-pair 0..7
            int bg = tid & 15;       // 0..15 (float4 groups across BT=64)
            const float* pc = Tc + (size_t)(k0 + 2 * kp) * TD + nBlock + bg * 4;
            const float* ps = Ts + (size_t)(k0 + 2 * kp) * TD + nBlock + bg * 4;
            store_pairs(Bc + kp * LDBP + bg * 8, *(const float4*)pc, *(const float4*)(pc + TD));
            store_pairs(Bs + kp * LDBP + bg * 8, *(const float4*)ps, *(const float4*)(ps + TD));
        }
        if (k0 + KT < TD) {  // pull next tiles toward the caches
            __builtin_prefetch(Ab + (size_t)(mBlock + (tid >> 2)) * TD + k0 + KT, 0, 1);
            __builtin_prefetch(Tc + (size_t)(k0 + KT + (tid >> 4)) * TD + nBlock, 0, 1);
            __builtin_prefetch(Ts + (size_t)(k0 + KT + (tid >> 4)) * TD + nBlock, 0, 1);
        }
        __syncthreads();

        // ---- 4 WMMA k-substeps of K=4 ----
#pragma unroll
        for (int kk = 0; kk < KT; kk += 4) {
            const int ka = kk + 2 * lhi;        // A k-offset for this half-wave
            const int kp = (kk >> 1) + lhi;     // B pair row
            v2f aF[2];
            aF[0] = *(const v2f*)(As + (waveM + lrow) * LDA + ka);
            aF[1] = *(const v2f*)(As + (waveM + 16 + lrow) * LDA + ka);
            v2f bFc[2], bFs[2];
#pragma unroll
            for (int j = 0; j < 2; ++j) {
                int nc = waveN + j * 16 + lrow;
                bFc[j] = *(const v2f*)(Bc + kp * LDBP + 2 * nc);
                bFs[j] = *(const v2f*)(Bs + kp * LDBP + 2 * nc);
            }
#pragma unroll
            for (int i = 0; i < 2; ++i)
#pragma unroll
                for (int j = 0; j < 2; ++j) {
                    accR[i][j] = wmma_f32(aF[i], bFc[j], accR[i][j]);
                    accI[i][j] = wmma_f32(aF[i], bFs[j], accI[i][j]);
                }
        }
        __syncthreads();
    }

    // ---- store: D layout M = v + 8*lhi, N = lrow within each 16x16 ----
    const size_t ob = (size_t)b * TDTD;
#pragma unroll
    for (int i = 0; i < 2; ++i)
#pragma unroll
        for (int j = 0; j < 2; ++j) {
            int mB = mBlock + waveM + i * 16 + 8 * lhi;
            int nC = nBlock + waveN + j * 16 + lrow;
#pragma unroll
            for (int v = 0; v < 8; ++v) {
                Xr[ob + (size_t)(mB + v) * TD + nC] = accR[i][j][v];
                Xi[ob + (size_t)(mB + v) * TD + nC] = accI[i][j][v];
            }
        }
}

// ---------------------------------------------------------------------------
// 6) Stage 2: out[b] = OUT_SCALE * (Tc @ Xr[b] - Ts @ Xi[b])   (contract rows)
//    f32 WMMA has no A/B NEG -> stage -Ts into LDS once per tile instead.
// ---------------------------------------------------------------------------
__global__ __launch_bounds__(128) void gemm_stage2(const float* __restrict__ Tc,
                                                   const float* __restrict__ Ts,
                                                   const float* __restrict__ Xr,
                                                   const float* __restrict__ Xi,
                                                   float* __restrict__ out) {
    __shared__ float AsC[BT * LDA];
    __shared__ float AsS[BT * LDA];                 // holds -Ts
    __shared__ float Br[(KT / 2) * LDBP];
    __shared__ float Bi[(KT / 2) * LDBP];

    const int b      = blockIdx.z;
    const int mBlock = blockIdx.y * BT;
    const int nBlock = blockIdx.x * BT;
    const float* XrB = Xr + (size_t)b * TDTD;
    const float* XiB = Xi + (size_t)b * TDTD;

    const int tid   = threadIdx.x;
    const int wave  = tid >> 5;
    const int lane  = tid & 31;
    const int waveM = (wave >> 1) * 32;
    const int waveN = (wave & 1) * 32;
    const int lrow  = lane & 15;
    const int lhi   = lane >> 4;

    v8f acc[2][2];
#pragma unroll
    for (int i = 0; i < 2; ++i)
#pragma unroll
        for (int j = 0; j < 2; ++j) acc[i][j] = (v8f)(0.0f);

    for (int k0 = 0; k0 < TD; k0 += KT) {
        {
            int r  = tid >> 2;
            int cg = tid & 3;
#pragma unroll
            for (int h = 0; h < 2; ++h) {
                int rr = r + 32 * h;
                float4 cv = *(const float4*)(Tc + (size_t)(mBlock + rr) * TD + k0 + cg * 4);
                float4 sv = *(const float4*)(Ts + (size_t)(mBlock + rr) * TD + k0 + cg * 4);
                sv.x = -sv.x; sv.y = -sv.y; sv.z = -sv.z; sv.w = -sv.w;   // negate once here
                *(float4*)(AsC + rr * LDA + cg * 4) = cv;
                *(float4*)(AsS + rr * LDA + cg * 4) = sv;
            }
            int kp = tid >> 4;
            int bg = tid & 15;
            const float* pr = XrB + (size_t)(k0 + 2 * kp) * TD + nBlock + bg * 4;
            const float* pi = XiB + (size_t)(k0 + 2 * kp) * TD + nBlock + bg * 4;
            store_pairs(Br + kp * LDBP + bg * 8, *(const float4*)pr, *(const float4*)(pr + TD));
            store_pairs(Bi + kp * LDBP + bg * 8, *(const float4*)pi, *(const float4*)(pi + TD));
        }
        if (k0 + KT < TD) {
            __builtin_prefetch(XrB + (size_t)(k0 + KT + (tid >> 4)) * TD + nBlock, 0, 1);
            __builtin_prefetch(XiB + (size_t)(k0 + KT + (tid >> 4)) * TD + nBlock, 0, 1);
        }
        __syncthreads();

#pragma unroll
        for (int kk = 0; kk < KT; kk += 4) {
            const int ka = kk + 2 * lhi;
            const int kp = (kk >> 1) + lhi;
            v2f aC[2], aSn[2];
            aC[0]  = *(const v2f*)(AsC + (waveM + lrow) * LDA + ka);
            aC[1]  = *(const v2f*)(AsC + (waveM + 16 + lrow) * LDA + ka);
            aSn[0] = *(const v2f*)(AsS + (waveM + lrow) * LDA + ka);
            aSn[1] = *(const v2f*)(AsS + (waveM + 16 + lrow) * LDA + ka);
            v2f bR[2], bI[2];
#pragma unroll
            for (int j = 0; j < 2; ++j) {
                int nc = waveN + j * 16 + lrow;
                bR[j] = *(const v2f*)(Br + kp * LDBP + 2 * nc);
                bI[j] = *(const v2f*)(Bi + kp * LDBP + 2 * nc);
            }
#pragma unroll
            for (int i = 0; i < 2; ++i)
#pragma unroll
                for (int j = 0; j < 2; ++j) {
                    acc[i][j] = wmma_f32(aC[i],  bR[j], acc[i][j]);
                    acc[i][j] = wmma_f32(aSn[i], bI[j], acc[i][j]);
                }
        }
        __syncthreads();
    }

    const size_t ob = (size_t)b * TDTD;
#pragma unroll
    for (int i = 0; i < 2; ++i)
#pragma unroll
        for (int j = 0; j < 2; ++j) {
            int mB = mBlock + waveM + i * 16 + 8 * lhi;
            int nC = nBlock + waveN + j * 16 + lrow;
#pragma unroll
            for (int v = 0; v < 8; ++v)
                out[ob + (size_t)(mB + v) * TD + nC] = acc[i][j][v] * OUT_SCALE;
        }
}

// ---------------------------------------------------------------------------
// Launch
// ---------------------------------------------------------------------------
extern "C" void kernel_launch(void* const* d_in, const int* in_sizes, int n_in,
                              void* d_out, int out_size, void* d_ws, size_t ws_size,
                              hipStream_t stream) {
    const float* cls   = (const float*)d_in[0];
    const float* W     = (const float*)d_in[1];
    const float* bias  = (const float*)d_in[2];
    const float* coeff = (const float*)d_in[3];
    const int*   lidx  = (const int*)d_in[4];
    float* out = (float*)d_out;

    char* ws = (char*)d_ws;
    size_t off = 0;
    float* Tc   = (float*)(ws + off); off += (size_t)TDTD * 4;            // 2.25 MB
    float* Ts   = (float*)(ws + off); off += (size_t)TDTD * 4;            // 2.25 MB
    int*   sel  = (int*)  (ws + off); off += 256;
    float* selw = (float*)(ws + off); off += 256;
    float* mask = (float*)(ws + off); off += (size_t)NB * TDTD * 4;       // 75.5 MB
    float* Xr   = (float*)(ws + off); off += (size_t)NB * TDTD * 4;       // 75.5 MB
    float* Xi   = (float*)(ws + off); off += (size_t)NB * TDTD * 4;       // 75.5 MB

    // 1) twiddle tables
    twiddle_kernel<<<(TDTD + 255) / 256, 256, 0, stream>>>(Tc, Ts);
    // 2) router + top-2 weights
    router_kernel<<<NB, NE, 0, stream>>>(cls, W, bias, sel, selw);
    // 3) zero the scatter grid (every call: graph replays must not accumulate)
    long long n4 = (long long)NB * TDTD / 4;
    zero_kernel<<<2048, 256, 0, stream>>>((float4*)mask, n4);
    // 4) scatter selected coefficients (indices globally unique -> plain store)
    scatter_kernel<<<(NB * 2 * NF) / 256, 256, 0, stream>>>(coeff, lidx, sel, selw, mask);
    // 5) row-wise IDFT as GEMM: X = mask @ (Tc + i Ts)
    dim3 g(TD / BT, TD / BT, NB);
    gemm_stage1<<<g, 128, 0, stream>>>(mask, Tc, Ts, Xr, Xi);
    // 6) column-wise IDFT + real part: out = scale*(Tc@Xr - Ts@Xi)
    gemm_stage2<<<g, 128, 0, stream>>>(Tc, Ts, Xr, Xi, out);
}